// HFDNN_68874095559014
// MI455X (gfx1250) — compile-verified
//
#include <hip/hip_runtime.h>
#include <cstddef>

typedef __attribute__((ext_vector_type(2))) float v2f;
typedef __attribute__((ext_vector_type(8))) float v8f;

#define LRELU(v) fmaxf((v), 0.01f * (v))

constexpr int NBLK  = 1024;   // blocks for layer kernels
constexpr int TPB   = 256;    // 8 waves (wave32)
constexpr int WAVES = NBLK * TPB / 32;

// One pass of: [optional BN+LReLU of previous z] -> 6x6 linear via WMMA ->
// write z planes (column-major [6][B]) + deterministic per-block stat partials.
__global__ void __launch_bounds__(TPB)
k_layer(const float* __restrict__ x, const float* __restrict__ Fh,
        const float* __restrict__ Ws, const float* __restrict__ bs,
        float* __restrict__ zp, float* __restrict__ partials,
        const float* __restrict__ stats, int B, int layer)
{
    const int lane  = threadIdx.x & 31;
    const int wid   = threadIdx.x >> 5;
    const int gwave = (blockIdx.x * TPB + threadIdx.x) >> 5;
    const bool lo   = lane < 16;
    const int  m    = lane & 15;
    const int  nTiles = B >> 4;
    const size_t Bs = (size_t)B;

    // ---- wave-uniform parameters
    const float* Wl = Ws + layer * 36;
    const float* bl = bs + layer * 6;
    float fw0=0,fw1=0,fw2=0,fw3=0,fw4=0,fw5=0;
    if (layer == 0) {
        fw0 = Fh[0]; fw1 = Fh[1]; fw2 = Fh[2];
        fw3 = Fh[3]; fw4 = Fh[4]; fw5 = Fh[5];
    }
    float sc[6], sh[6];
    if (layer > 0) {
        const float* st = stats + (layer - 1) * 12;
        #pragma unroll
        for (int j = 0; j < 6; ++j) { sc[j] = st[j]; sh[j] = st[6 + j]; }
    }

    // ---- A operands: W as 16x4 f32 A-matrix (M=feature padded to 16 with 0)
    // A layout: lanes 0-15 M=lane (VGPR0 K=0, VGPR1 K=1); lanes 16-31 K=2,3.
    const int  mm = (m < 6) ? m : 0;
    const int  kb = lo ? 0 : 2;
    const float w0 = Wl[mm * 6 + kb];
    const float w1 = Wl[mm * 6 + kb + 1];
    const float w4 = Wl[mm * 6 + 4];
    const float w5 = Wl[mm * 6 + 5];
    const bool mv = (m < 6);
    v2f a0, a1;
    a0.x = mv ? w0 : 0.0f;
    a0.y = mv ? w1 : 0.0f;
    a1.x = (mv && lo) ? w4 : 0.0f;   // K=4,5 in lower half; K=6,7 pad = 0
    a1.y = (mv && lo) ? w5 : 0.0f;

    // ---- C operand: bias on M=0..5 (vgpr r, lower lanes), zero elsewhere
    v8f cb;
    #pragma unroll
    for (int r = 0; r < 8; ++r) {
        float bb = (r < 6) ? bl[r] : 0.0f;
        cb[r] = (lo && r < 6) ? bb : 0.0f;
    }

    float s0=0,s1=0,s2=0,s3=0,s4=0,s5=0;      // per-feature sum partials
    float q0=0,q1=0,q2=0,q3=0,q4=0,q5=0;      // per-feature sumsq partials

    for (int tile = gwave; tile < nTiles; tile += WAVES) {
        const int row = (tile << 4) + m;

        int nxt = tile + WAVES;
        if (nxt < nTiles) {                    // CDNA5 global_prefetch path
            if (layer == 0) __builtin_prefetch(x + (size_t)((nxt << 4) + m) * 3, 0, 0);
            else            __builtin_prefetch(zp + ((nxt << 4) + m), 0, 0);
        }

        float h0, h1, h2, h3, h4, h5;
        if (layer == 0) {
            // front end: 3->2 linear, L2 norm, polynomial features v^0..v^5
            const float x0 = x[(size_t)row * 3 + 0];
            const float x1 = x[(size_t)row * 3 + 1];
            const float x2 = x[(size_t)row * 3 + 2];
            const float e0 = fw0 * x0 + fw1 * x1 + fw2 * x2;
            const float e1 = fw3 * x0 + fw4 * x1 + fw5 * x2;
            const float v  = sqrtf(e0 * e0 + e1 * e1);
            h0 = 1.0f; h1 = v; h2 = v * v; h3 = h2 * v; h4 = h2 * h2; h5 = h4 * v;
        } else {
            // lazy BN+LReLU of previous layer's pre-BN z
            float t0 = zp[0 * Bs + row], t1 = zp[1 * Bs + row], t2 = zp[2 * Bs + row];
            float t3 = zp[3 * Bs + row], t4 = zp[4 * Bs + row], t5 = zp[5 * Bs + row];
            h0 = LRELU(fmaf(t0, sc[0], sh[0]));
            h1 = LRELU(fmaf(t1, sc[1], sh[1]));
            h2 = LRELU(fmaf(t2, sc[2], sh[2]));
            h3 = LRELU(fmaf(t3, sc[3], sh[3]));
            h4 = LRELU(fmaf(t4, sc[4], sh[4]));
            h5 = LRELU(fmaf(t5, sc[5], sh[5]));
        }

        // B operands: 4x16 f32 B-matrix; lanes 0-15 K=0,1 / lanes 16-31 K=2,3
        v2f b0, b1;
        b0.x = lo ? h0 : h2;
        b0.y = lo ? h1 : h3;
        b1.x = lo ? h4 : 0.0f;   // K=6,7 pad rows = 0
        b1.y = lo ? h5 : 0.0f;

        // z[feature][row] tile = W * h + b   (exact f32 WMMA, K = 4 + 2)
        v8f d = __builtin_amdgcn_wmma_f32_16x16x4_f32(
            false, a0, false, b0, (short)0, cb, false, false);
        d = __builtin_amdgcn_wmma_f32_16x16x4_f32(
            false, a1, false, b1, (short)0, d, false, false);

        const float d0 = d[0], d1 = d[1], d2 = d[2];
        const float d3 = d[3], d4 = d[4], d5 = d[5];

        // upper-lane D values are exactly 0 (A rows >=6 and upper C are 0),
        // so all 32 lanes may accumulate.
        s0 += d0; q0 += d0 * d0;   s1 += d1; q1 += d1 * d1;
        s2 += d2; q2 += d2 * d2;   s3 += d3; q3 += d3 * d3;
        s4 += d4; q4 += d4 * d4;   s5 += d5; q5 += d5 * d5;

        if (lo) {   // lane n owns row n of the tile: 6 coalesced plane stores
            zp[0 * Bs + row] = d0;  zp[1 * Bs + row] = d1;
            zp[2 * Bs + row] = d2;  zp[3 * Bs + row] = d3;
            zp[4 * Bs + row] = d4;  zp[5 * Bs + row] = d5;
        }
    }

    // ---- deterministic block reduction of stat partials
    float acc[12] = { s0,s1,s2,s3,s4,s5, q0,q1,q2,q3,q4,q5 };
    #pragma unroll
    for (int j = 0; j < 12; ++j) {
        #pragma unroll
        for (int off = 16; off > 0; off >>= 1)
            acc[j] += __shfl_xor(acc[j], off, 32);
    }
    __shared__ float red[TPB / 32][12];
    if (lane == 0) {
        #pragma unroll
        for (int j = 0; j < 12; ++j) red[wid][j] = acc[j];
    }
    __syncthreads();
    if (threadIdx.x < 12) {
        float t = 0.0f;
        #pragma unroll
        for (int w = 0; w < TPB / 32; ++w) t += red[w][threadIdx.x];
        partials[(size_t)blockIdx.x * 12 + threadIdx.x] = t;
    }
}

// Fold per-block partials -> mean/var -> (scale, shift) for one layer.
__global__ void k_stats(const float* __restrict__ partials,
                        const float* __restrict__ gammas,
                        const float* __restrict__ betas,
                        float* __restrict__ stats, int layer, float invB)
{
    __shared__ float sums[12];
    const int f = threadIdx.x;
    if (f < 12) {
        double t = 0.0;                       // fixed-order, deterministic
        for (int b = 0; b < NBLK; ++b) t += (double)partials[(size_t)b * 12 + f];
        sums[f] = (float)t;
    }
    __syncthreads();
    if (f < 6) {
        const float mean = sums[f] * invB;
        float var = fmaf(-mean, mean, sums[6 + f] * invB);
        var = fmaxf(var, 0.0f);
        const float scale = gammas[layer * 6 + f] * rsqrtf(var + 1e-5f);
        stats[layer * 12 + f]     = scale;
        stats[layer * 12 + 6 + f] = betas[layer * 6 + f] - mean * scale;
    }
}

// Final: BN+LReLU of layer-8 z, 6->1 decode, LReLU, write out.
__global__ void __launch_bounds__(256)
k_final(const float* __restrict__ zp, const float* __restrict__ stats,
        const float* __restrict__ dw, const float* __restrict__ db,
        float* __restrict__ out, int B)
{
    const size_t Bs = (size_t)B;
    float sc[6], sh[6], w[6];
    #pragma unroll
    for (int j = 0; j < 6; ++j) {
        sc[j] = stats[7 * 12 + j];
        sh[j] = stats[7 * 12 + 6 + j];
        w[j]  = dw[j];
    }
    const float bias = db[0];
    const int stride = gridDim.x * blockDim.x;
    for (int row = blockIdx.x * blockDim.x + threadIdx.x; row < B; row += stride) {
        float acc = bias;
        #pragma unroll
        for (int f = 0; f < 6; ++f) {
            const float h = LRELU(fmaf(zp[(size_t)f * Bs + row], sc[f], sh[f]));
            acc = fmaf(w[f], h, acc);
        }
        out[row] = LRELU(acc);
    }
}

extern "C" void kernel_launch(void* const* d_in, const int* in_sizes, int n_in,
                              void* d_out, int out_size, void* d_ws, size_t ws_size,
                              hipStream_t stream)
{
    const float* x   = (const float*)d_in[0];   // [B,3]
    const float* Fh  = (const float*)d_in[1];   // [2,3]
    const float* Ws  = (const float*)d_in[2];   // [8,6,6]
    const float* bs  = (const float*)d_in[3];   // [8,6]
    const float* gam = (const float*)d_in[4];   // [8,6]
    const float* bet = (const float*)d_in[5];   // [8,6]
    const float* dw  = (const float*)d_in[6];   // [1,6]
    const float* db  = (const float*)d_in[7];   // [1]
    const int B = in_sizes[0] / 3;

    float* zp       = (float*)d_ws;                 // 6 * B floats (z planes)
    float* partials = zp + (size_t)6 * B;           // NBLK * 12 floats
    float* stats    = partials + (size_t)NBLK * 12; // 8 * 12 floats

    const float invB = 1.0f / (float)B;
    for (int layer = 0; layer < 8; ++layer) {
        k_layer<<<NBLK, TPB, 0, stream>>>(x, Fh, Ws, bs, zp, partials, stats, B, layer);
        k_stats<<<1, 64, 0, stream>>>(partials, gam, bet, stats, layer, invB);
    }
    k_final<<<2048, 256, 0, stream>>>(zp, stats, dw, db, (float*)d_out, B);
}